// GCNAnomalyDetector_15126874816802
// MI455X (gfx1250) — compile-verified
//
#include <hip/hip_runtime.h>

#define NN 50000
#define EE 800000
#define FF 128
#define HH 128

typedef float v2f __attribute__((ext_vector_type(2)));
typedef float v8f __attribute__((ext_vector_type(8)));

__global__ void zero_kernel(float* __restrict__ p, int n) {
  int i = blockIdx.x * blockDim.x + threadIdx.x;
  if (i < n) p[i] = 0.0f;
}

__global__ void deg_kernel(const int* __restrict__ dst, float* __restrict__ deg, int ne) {
  int e = blockIdx.x * blockDim.x + threadIdx.x;
  if (e < ne) atomicAdd(&deg[dst[e]], 1.0f);
}

__global__ void dinv_kernel(float* __restrict__ deg, int n) {
  int i = blockIdx.x * blockDim.x + threadIdx.x;
  if (i < n) deg[i] = rsqrtf(deg[i] + 1.0f);  // +1 for self loop; always > 0
}

// C[M x 128] = A[M x 128] @ W[128 x 128], fp32 WMMA, one 16x16 tile per wave.
__global__ void gemm_wmma(const float* __restrict__ A, const float* __restrict__ W,
                          float* __restrict__ C, int M) {
  const int K = 128, NCOL = 128;
  int wave = (blockIdx.x * blockDim.x + threadIdx.x) >> 5;
  int lane = threadIdx.x & 31;
  const int tilesN = NCOL / 16;  // 8
  int tm = wave / tilesN;
  int tn = wave - tm * tilesN;
  if (tm * 16 >= M) return;      // uniform per-wave: EXEC stays all-ones
  int row0 = tm * 16, col0 = tn * 16;
  int half = lane >> 4;          // 0: K=k,k+1  1: K=k+2,k+3  (ISA f32 A layout)
  int l = lane & 15;
  const float* Arow = A + (size_t)(row0 + l) * K;
  const float* Wcol = W + col0 + l;
  v8f acc = {};
#pragma unroll 8
  for (int k = 0; k < K; k += 4) {
    int ka = k + half * 2;
    v2f a, b;
    a[0] = Arow[ka];
    a[1] = Arow[ka + 1];
    b[0] = Wcol[(size_t)ka * NCOL];
    b[1] = Wcol[(size_t)(ka + 1) * NCOL];
    acc = __builtin_amdgcn_wmma_f32_16x16x4_f32(
        /*neg_a=*/false, a, /*neg_b=*/false, b,
        /*c_mod=*/(short)0, acc, /*reuse_a=*/false, /*reuse_b=*/false);
  }
  // C/D layout: VGPR r -> row (row0 + r + half*8), col (col0 + l)
  float* Cc = C + col0 + l;
#pragma unroll
  for (int r = 0; r < 8; ++r)
    Cc[(size_t)(row0 + r + half * 8) * NCOL] = acc[r];
}

// One wave per edge; each lane handles 4 consecutive features (coalesced 512B row).
__global__ void scatter_kernel(const float* __restrict__ xw, const int* __restrict__ src,
                               const int* __restrict__ dst, const float* __restrict__ dinv,
                               float* __restrict__ acc, int ne) {
  int idx = blockIdx.x * blockDim.x + threadIdx.x;
  int e = idx >> 5;
  if (e >= ne) return;
  int j = (idx & 31) * 4;
  int s = src[e], d = dst[e];
  float norm = dinv[s] * dinv[d];
  const float4 v = *(const float4*)(xw + (size_t)s * HH + j);
  float* o = acc + (size_t)d * HH + j;
  atomicAdd(o + 0, v.x * norm);
  atomicAdd(o + 1, v.y * norm);
  atomicAdd(o + 2, v.z * norm);
  atomicAdd(o + 3, v.w * norm);
}

// Fused self-loop contribution + bias + ReLU (in place on acc).
__global__ void finalize_kernel(const float* __restrict__ xw, const float* __restrict__ dinv,
                                const float* __restrict__ bias, float* __restrict__ acc, int n) {
  int idx = blockIdx.x * blockDim.x + threadIdx.x;
  int i = idx >> 5;
  if (i >= n) return;
  int j = (idx & 31) * 4;
  float di = dinv[i];
  float w = di * di;
  const float4 v = *(const float4*)(xw + (size_t)i * HH + j);
  const float4 bb = *(const float4*)(bias + j);
  float4 a = *(float4*)(acc + (size_t)i * HH + j);
  a.x = fmaxf(fmaf(v.x, w, a.x) + bb.x, 0.0f);
  a.y = fmaxf(fmaf(v.y, w, a.y) + bb.y, 0.0f);
  a.z = fmaxf(fmaf(v.z, w, a.z) + bb.z, 0.0f);
  a.w = fmaxf(fmaf(v.w, w, a.w) + bb.w, 0.0f);
  *(float4*)(acc + (size_t)i * HH + j) = a;
}

// out[i] = dot(h[i,:], Wf) + bf  — one wave per node, shfl_xor reduction (wave32).
__global__ void head_kernel(const float* __restrict__ h, const float* __restrict__ Wf,
                            const float* __restrict__ bf, float* __restrict__ out, int n) {
  int idx = blockIdx.x * blockDim.x + threadIdx.x;
  int node = idx >> 5;
  int lane = idx & 31;
  if (node >= n) return;
  const float* hr = h + (size_t)node * HH;
  float s = 0.0f;
#pragma unroll
  for (int j = 0; j < HH / 32; ++j) s = fmaf(hr[lane + j * 32], Wf[lane + j * 32], s);
#pragma unroll
  for (int off = 16; off > 0; off >>= 1) s += __shfl_xor(s, off, 32);
  if (lane == 0) out[node] = s + bf[0];
}

extern "C" void kernel_launch(void* const* d_in, const int* in_sizes, int n_in,
                              void* d_out, int out_size, void* d_ws, size_t ws_size,
                              hipStream_t stream) {
  const float* x  = (const float*)d_in[0];
  const int*   ei = (const int*)d_in[1];
  const float* W1 = (const float*)d_in[2];
  const float* b1 = (const float*)d_in[3];
  const float* W2 = (const float*)d_in[4];
  const float* b2 = (const float*)d_in[5];
  const float* Wf = (const float*)d_in[6];
  const float* bf = (const float*)d_in[7];
  float* out = (float*)d_out;

  const int* srcp = ei;        // edge_index[0]
  const int* dstp = ei + EE;   // edge_index[1]

  float* ws   = (float*)d_ws;
  float* dinv = ws;                          // N floats
  float* buf0 = ws + NN;                     // N*H floats (xw)
  float* buf1 = buf0 + (size_t)NN * HH;      // N*H floats (acc / h)

  const int NH = NN * HH;
  dim3 b256(256);
  int gN   = (NN + 255) / 256;
  int gE   = (EE + 255) / 256;
  int gNH  = (NH + 255) / 256;
  int gE32 = (EE * 32) / 256;                 // exact: 100000
  int gN32 = (NN * 32 + 255) / 256;
  int gGemm = ((NN / 16) * (HH / 16) * 32) / 128;  // 25000 waves / 4 per block

  // normalization: dinv = rsqrt(indeg + 1)
  zero_kernel<<<gN, b256, 0, stream>>>(dinv, NN);
  deg_kernel<<<gE, b256, 0, stream>>>(dstp, dinv, EE);
  dinv_kernel<<<gN, b256, 0, stream>>>(dinv, NN);

  // layer 1: xw1 = x@W1 -> buf0 ; propagate -> buf1 ; h1 = relu(...+b1) in buf1
  gemm_wmma<<<gGemm, 128, 0, stream>>>(x, W1, buf0, NN);
  zero_kernel<<<gNH, b256, 0, stream>>>(buf1, NH);
  scatter_kernel<<<gE32, b256, 0, stream>>>(buf0, srcp, dstp, dinv, buf1, EE);
  finalize_kernel<<<gN32, b256, 0, stream>>>(buf0, dinv, b1, buf1, NN);

  // layer 2: xw2 = h1@W2 -> buf0 (h1 consumed) ; acc reuses buf1
  gemm_wmma<<<gGemm, 128, 0, stream>>>(buf1, W2, buf0, NN);
  zero_kernel<<<gNH, b256, 0, stream>>>(buf1, NH);
  scatter_kernel<<<gE32, b256, 0, stream>>>(buf0, srcp, dstp, dinv, buf1, EE);
  finalize_kernel<<<gN32, b256, 0, stream>>>(buf0, dinv, b2, buf1, NN);

  // head
  head_kernel<<<gN32, b256, 0, stream>>>(buf1, Wf, bf, out, NN);
}